// TaskAlignedAssigner_21517786153047
// MI455X (gfx1250) — compile-verified
//
#include <hip/hip_runtime.h>
#include <math.h>
#include <stdint.h>

#define BS   16
#define NA   8400
#define NM   64
#define NC   80
#define KTOP 10
#define EPSF 1e-9f
#define IOU_EPSF 1e-7f

typedef float v8f __attribute__((ext_vector_type(8)));
typedef float v2f __attribute__((ext_vector_type(2)));

// ---------- CDNA5 async global->LDS helpers (ASYNCcnt path, ISA ch.10/15.18) ----
__device__ __forceinline__ void async_load_b128(uint32_t lds_off, const void* gaddr) {
    asm volatile("global_load_async_to_lds_b128 %0, %1, off"
                 :: "v"(lds_off), "v"(gaddr) : "memory");
}
__device__ __forceinline__ void wait_async0() {
#if __has_builtin(__builtin_amdgcn_s_wait_asynccnt)
    __builtin_amdgcn_s_wait_asynccnt(0);
#else
    asm volatile("s_wait_asynccnt 0" ::: "memory");
#endif
}

// ---------- CIoU (box1 = gt, box2 = pd), clipped to >= 0 ------------------------
__device__ __forceinline__ float ciou_clip(float4 g, float4 p) {
    const float eps = IOU_EPSF;
    float w1 = g.z - g.x, h1 = g.w - g.y + eps;
    float w2 = p.z - p.x, h2 = p.w - p.y + eps;
    float iw = fmaxf(fminf(g.z, p.z) - fmaxf(g.x, p.x), 0.0f);
    float ih = fmaxf(fminf(g.w, p.w) - fmaxf(g.y, p.y), 0.0f);
    float inter = iw * ih;
    float uni = w1 * h1 + w2 * h2 - inter + eps;
    float iou = inter / uni;
    float cw = fmaxf(g.z, p.z) - fminf(g.x, p.x);
    float ch = fmaxf(g.w, p.w) - fminf(g.y, p.y);
    float c2 = cw * cw + ch * ch + eps;
    float dx = (p.x + p.z) - (g.x + g.z);
    float dy = (p.y + p.w) - (g.y + g.w);
    float rho2 = (dx * dx + dy * dy) * 0.25f;
    float dth = atanf(w2 / h2) - atanf(w1 / h1);
    float v = 0.40528473456935108578f * dth * dth;   // 4/pi^2
    float alpha = v / (v - iou + (1.0f + eps));
    float ciou = iou - (rho2 / c2 + v * alpha);
    return fmaxf(ciou, 0.0f);
}

// ---------- Kernel 1: bbox_scores[b,m,a] = pd_scores[b,a,label[b,m]] via WMMA ---
// D(16 anchors x 16 gts) = A(16 anchors x 4 classes) x B(one-hot 4 x 16) over 20 K-steps.
// Exact in f32 (multiplies by 1.0/0.0, adds zeros).
__global__ void gather_scores_wmma(const float* __restrict__ pd_scores,
                                   const int* __restrict__ gt_labels,
                                   float* __restrict__ bbox_scores) {
    int w    = (blockIdx.x * blockDim.x + threadIdx.x) >> 5;
    int lane = threadIdx.x & 31;
    const int ATILES = NA / 16, GTILES = NM / 16;
    if (w >= BS * ATILES * GTILES) return;     // uniform per wave
    int b     = w / (ATILES * GTILES);
    int rem   = w % (ATILES * GTILES);
    int atile = rem / GTILES;
    int gg    = rem % GTILES;
    int n     = lane & 15;        // column (gt) for B/D; row (anchor) for A
    int half  = lane >> 4;        // K-half selector (K pairs 0,1 vs 2,3)
    int abase = atile * 16;
    int gt    = gg * 16 + n;
    int lab   = gt_labels[b * NM + gt];
#if __has_builtin(__builtin_amdgcn_wmma_f32_16x16x4_f32)
    v8f acc = {};
    const float* arow = pd_scores + ((size_t)(b * NA + abase + n)) * NC;
    #pragma unroll
    for (int k0 = 0; k0 < NC; k0 += 4) {
        int kk = k0 + 2 * half;
        v2f a;  a.x = arow[kk];  a.y = arow[kk + 1];
        v2f bb; bb.x = (lab == kk)     ? 1.0f : 0.0f;
                bb.y = (lab == kk + 1) ? 1.0f : 0.0f;
        acc = __builtin_amdgcn_wmma_f32_16x16x4_f32(false, a, false, bb,
                                                    (short)0, acc, false, false);
    }
    // D layout: lane<16 -> M=r (VGPR r), N=lane ; lane>=16 -> M=8+r, N=lane-16
    float* out = bbox_scores + (size_t)(b * NM + gt) * NA + abase + half * 8;
    *(float4*)(out)     = make_float4(acc[0], acc[1], acc[2], acc[3]);
    *(float4*)(out + 4) = make_float4(acc[4], acc[5], acc[6], acc[7]);
#else
    float* outrow = bbox_scores + (size_t)(b * NM + gt) * NA;
    int a0 = abase + half * 8;
    #pragma unroll
    for (int r = 0; r < 8; ++r)
        outrow[a0 + r] = pd_scores[((size_t)(b * NA + a0 + r)) * NC + lab];
#endif
}

// ---------- Kernel 2: init scratch ---------------------------------------------
__global__ void init_kernel(unsigned* count, int* min_m,
                            unsigned* pos_align, unsigned* pos_ov) {
    int t = blockIdx.x * blockDim.x + threadIdx.x;
    if (t < BS * NA) { count[t] = 0u; min_m[t] = 0x7fffffff; }
    if (t < BS * NM) { pos_align[t] = 0u; pos_ov[t] = 0u; }
}

// ---------- Kernel 3: per-(b,m) wave: top-10 of align*in_gts with index tiebreak
__global__ void topk_kernel(const float4* __restrict__ pd_bboxes,
                            const float2* __restrict__ anc,
                            const float4* __restrict__ gt_bboxes,
                            const float* __restrict__ bbox_scores,
                            int* __restrict__ cand_idx) {
    int wid  = (blockIdx.x * blockDim.x + threadIdx.x) >> 5;
    int lane = threadIdx.x & 31;
    if (wid >= BS * NM) return;
    int b = wid >> 6;
    float4 g = gt_bboxes[wid];
    const float4* pb = pd_bboxes + b * NA;
    const float*  sc = bbox_scores + (size_t)wid * NA;

    unsigned long long K[KTOP];
    #pragma unroll
    for (int i = 0; i < KTOP; ++i) K[i] = 0ull;

    for (int a = lane; a < NA; a += 32) {
        float4 p = pb[a];
        float2 ap = anc[a];
        float dmin = fminf(fminf(ap.x - g.x, ap.y - g.y),
                           fminf(g.z - ap.x, g.w - ap.y));
        float ov = ciou_clip(g, p);
        float t3 = ov * ov * ov;
        float metric = sqrtf(sc[a]) * t3 * t3;
        float masked = (dmin > EPSF) ? metric : 0.0f;
        unsigned long long key =
            ((unsigned long long)__float_as_uint(masked) << 32) |
            (unsigned long long)(0xFFFFFFFFu - (unsigned)a);
        if (key > K[KTOP - 1]) {
            K[KTOP - 1] = key;
            #pragma unroll
            for (int i = KTOP - 1; i > 0; --i) {
                unsigned long long lo = K[i - 1] < K[i] ? K[i - 1] : K[i];
                unsigned long long hi = K[i - 1] < K[i] ? K[i] : K[i - 1];
                K[i - 1] = hi; K[i] = lo;
            }
        }
    }
    // merge 32 x 10 -> global top-10 (keys are unique: index embedded)
    unsigned long long chosen = 0ull, first = 0ull;
    #pragma unroll
    for (int t = 0; t < KTOP; ++t) {
        unsigned long long cand = K[0];
        unsigned long long v = cand;
        #pragma unroll
        for (int off = 16; off >= 1; off >>= 1) {
            unsigned long long o = __shfl_xor(v, off, 32);
            if (o > v) v = o;
        }
        if (t == 0) first = v;
        if (lane == t) chosen = v;
        unsigned long long bal = __ballot(cand == v);
        int owner = __ffsll(bal) - 1;
        if (lane == owner) {
            #pragma unroll
            for (int i = 0; i < KTOP - 1; ++i) K[i] = K[i + 1];
            K[KTOP - 1] = 0ull;
        }
    }
    if (lane < KTOP) {
        int idx = -1;
        float rowmax = __uint_as_float((unsigned)(first >> 32));
        if (rowmax > EPSF) {   // topk valid for this row
            unsigned ai = 0xFFFFFFFFu - (unsigned)(chosen & 0xFFFFFFFFull);
            float2 ap = anc[ai];
            float dmin = fminf(fminf(ap.x - g.x, ap.y - g.y),
                               fminf(g.z - ap.x, g.w - ap.y));
            if (dmin > EPSF) idx = (int)ai;   // mask_pos = topk & in_gts
        }
        cand_idx[wid * KTOP + lane] = idx;
    }
}

// ---------- Kernel 4: scatter candidate counts / lowest-m per anchor -----------
__global__ void scatter_kernel(const int* __restrict__ cand_idx,
                               unsigned* count, int* min_m) {
    int t = blockIdx.x * blockDim.x + threadIdx.x;
    if (t >= BS * NM * KTOP) return;
    int idx = cand_idx[t];
    if (idx >= 0) {
        int s = t / KTOP;
        int b = s >> 6, m = s & 63;
        atomicAdd(&count[b * NA + idx], 1u);
        atomicMin(&min_m[b * NA + idx], m);
    }
}

// ---------- Kernel 5: per-anchor conflict resolution ----------------------------
// GT boxes staged to LDS with async global->LDS (ASYNCcnt) + s_wait_asynccnt.
__global__ void resolve_kernel(const float4* __restrict__ pd_bboxes,
                               const float4* __restrict__ gt_bboxes,
                               const float* __restrict__ bbox_scores,
                               const unsigned* __restrict__ count,
                               const int* __restrict__ min_m,
                               int* __restrict__ chosen_m,
                               float* __restrict__ chosen_met,
                               unsigned* pos_align, unsigned* pos_ov) {
    __shared__ float4 sGT[NM];
    int b = blockIdx.y;
    int tid = threadIdx.x;
    if (tid < NM) {
        uint32_t lds = (uint32_t)(uintptr_t)(&sGT[tid]);   // low 32 bits = LDS offset
        async_load_b128(lds, (const void*)(gt_bboxes + b * NM + tid));
        wait_async0();
    }
    __syncthreads();

    int a = blockIdx.x * blockDim.x + tid;
    if (a >= NA) return;
    int ba = b * NA + a;
    unsigned c = count[ba];
    int m = -1; float met = 0.0f;
    if (c == 1u) {
        m = min_m[ba];
        float4 p = pd_bboxes[ba];
        float ov = ciou_clip(sGT[m], p);
        float t3 = ov * ov * ov;
        met = sqrtf(bbox_scores[(size_t)(b * NM + m) * NA + a]) * t3 * t3;
        atomicMax(&pos_align[b * NM + m], __float_as_uint(met));
        atomicMax(&pos_ov[b * NM + m], __float_as_uint(ov));
    } else if (c > 1u) {
        // overlaps.argmax over all M (first max wins, matches jnp.argmax)
        float4 p = pd_bboxes[ba];
        float best = -1.0f; int bm = 0;
        #pragma unroll 4
        for (int mi = 0; mi < NM; ++mi) {
            float ov = ciou_clip(sGT[mi], p);
            if (ov > best) { best = ov; bm = mi; }
        }
        m = bm;
        float t3 = best * best * best;
        met = sqrtf(bbox_scores[(size_t)(b * NM + m) * NA + a]) * t3 * t3;
        atomicMax(&pos_align[b * NM + m], __float_as_uint(met));
        atomicMax(&pos_ov[b * NM + m], __float_as_uint(best));
    }
    chosen_m[ba] = m;
    chosen_met[ba] = met;
}

// ---------- Kernel 6: finalize labels / boxes / fg / gt_idx / norm --------------
__global__ void finalize_kernel(const float4* __restrict__ gt_bboxes,
                                const int* __restrict__ gt_labels,
                                const int* __restrict__ chosen_m,
                                const float* __restrict__ chosen_met,
                                const unsigned* __restrict__ pos_align,
                                const unsigned* __restrict__ pos_ov,
                                float* out_labels, float* out_bboxes,
                                float* out_fg, float* out_gtidx,
                                float* normw, int* labw) {
    int a = blockIdx.x * blockDim.x + threadIdx.x;
    int b = blockIdx.y;
    if (a >= NA) return;
    int ba = b * NA + a;
    int m = chosen_m[ba];
    int fg = (m >= 0);
    int tgt = fg ? m : 0;
    int lab = gt_labels[b * NM + tgt];
    float4 box = gt_bboxes[b * NM + tgt];
    float norm = 0.0f;
    if (fg) {
        float pa = __uint_as_float(pos_align[b * NM + m]);
        float po = __uint_as_float(pos_ov[b * NM + m]);
        norm = chosen_met[ba] * po / (pa + EPSF);
    }
    out_labels[ba] = (float)lab;
    ((float4*)out_bboxes)[ba] = box;
    out_fg[ba] = fg ? 1.0f : 0.0f;
    out_gtidx[ba] = (float)tgt;
    normw[ba] = norm;
    labw[ba] = lab;
}

// ---------- Kernel 7: one-hot target_scores * norm ------------------------------
__global__ void scores_kernel(const float* __restrict__ normw,
                              const int* __restrict__ labw,
                              float* __restrict__ out_scores) {
    int t = blockIdx.x * blockDim.x + threadIdx.x;
    if (t >= BS * NA * NC) return;
    int c = t % NC;
    int ba = t / NC;
    out_scores[t] = (c == labw[ba]) ? normw[ba] : 0.0f;
}

// -------------------------------------------------------------------------------
extern "C" void kernel_launch(void* const* d_in, const int* in_sizes, int n_in,
                              void* d_out, int out_size, void* d_ws, size_t ws_size,
                              hipStream_t stream) {
    (void)in_sizes; (void)n_in; (void)out_size; (void)ws_size;
    const float*  pd_scores = (const float*)d_in[0];
    const float4* pd_bboxes = (const float4*)d_in[1];
    const float2* anc       = (const float2*)d_in[2];
    const int*    gt_labels = (const int*)d_in[3];
    const float4* gt_bboxes = (const float4*)d_in[4];

    float* out = (float*)d_out;   // tuple concatenated flat, all as f32
    float* out_labels = out;                       // BS*NA
    float* out_bboxes = out + (size_t)BS * NA;     // BS*NA*4
    float* out_scores = out + (size_t)BS * NA * 5; // BS*NA*NC
    float* out_fg     = out + (size_t)BS * NA * (5 + NC);
    float* out_gtidx  = out + (size_t)BS * NA * (6 + NC);

    char* w = (char*)d_ws;
    auto alloc = [&](size_t bytes) { char* p = w; w += (bytes + 255) & ~(size_t)255; return p; };
    float*    bbox_scores = (float*)   alloc((size_t)BS * NM * NA * 4);
    int*      cand_idx    = (int*)     alloc((size_t)BS * NM * KTOP * 4);
    unsigned* count       = (unsigned*)alloc((size_t)BS * NA * 4);
    int*      min_m       = (int*)     alloc((size_t)BS * NA * 4);
    int*      chosen_m    = (int*)     alloc((size_t)BS * NA * 4);
    float*    chosen_met  = (float*)   alloc((size_t)BS * NA * 4);
    unsigned* pos_align   = (unsigned*)alloc((size_t)BS * NM * 4);
    unsigned* pos_ov      = (unsigned*)alloc((size_t)BS * NM * 4);
    float*    normw       = (float*)   alloc((size_t)BS * NA * 4);
    int*      labw        = (int*)     alloc((size_t)BS * NA * 4);

    // 1) WMMA one-hot gather: 16*(8400/16)*(64/16) = 33600 waves, 8 waves/block
    gather_scores_wmma<<<(BS * (NA / 16) * (NM / 16)) / 8, 256, 0, stream>>>(
        pd_scores, gt_labels, bbox_scores);
    // 2) init scratch
    init_kernel<<<(BS * NA + 255) / 256, 256, 0, stream>>>(count, min_m, pos_align, pos_ov);
    // 3) per-(b,m) wave top-10 -> sparse candidates
    topk_kernel<<<(BS * NM) / 8, 256, 0, stream>>>(pd_bboxes, anc, gt_bboxes,
                                                   bbox_scores, cand_idx);
    // 4) scatter counts / lowest-m
    scatter_kernel<<<(BS * NM * KTOP + 255) / 256, 256, 0, stream>>>(cand_idx, count, min_m);
    // 5) per-anchor resolve (async-LDS staged GT boxes)
    dim3 gBA((NA + 255) / 256, BS);
    resolve_kernel<<<gBA, 256, 0, stream>>>(pd_bboxes, gt_bboxes, bbox_scores,
                                            count, min_m, chosen_m, chosen_met,
                                            pos_align, pos_ov);
    // 6) finalize scalar outputs + norm
    finalize_kernel<<<gBA, 256, 0, stream>>>(gt_bboxes, gt_labels, chosen_m, chosen_met,
                                             pos_align, pos_ov, out_labels, out_bboxes,
                                             out_fg, out_gtidx, normw, labw);
    // 7) one-hot scores
    scores_kernel<<<(BS * NA * NC + 255) / 256, 256, 0, stream>>>(normw, labw, out_scores);
}